// SelfAttention_49667001811786
// MI455X (gfx1250) — compile-verified
//
#include <hip/hip_runtime.h>
#include <hip/hip_bf16.h>

// Problem constants from the reference
#define B_ 4
#define L_ 512
#define D_ 256
#define H_ 128

typedef float v2f __attribute__((ext_vector_type(2)));
typedef float v8f __attribute__((ext_vector_type(8)));

// D = A(16x4) * B(4x16) + C  in f32, wave32
static __device__ inline v8f wmma4(v2f a, v2f b, v8f c) {
    return __builtin_amdgcn_wmma_f32_16x16x4_f32(
        /*neg_a=*/false, a, /*neg_b=*/false, b,
        /*c_mod=*/(short)0, c, /*reuse_a=*/false, /*reuse_b=*/false);
}

static __device__ inline float sigmoidf_(float x) {
    return 1.0f / (1.0f + __expf(-x));
}

// ---------------------------------------------------------------------------
// f32 WMMA GEMM:  C[m,n] = sum_k A[m,k] * B(k,n) + bias[n]   (+ fused gating)
//   TRANSB=1 : B stored as W[N,K]  (row n contiguous in k)  -> C = A * W^T
//   TRANSB=0 : B stored as [K,N]
//   GATED    : C = sigmoid(lin) * Gsrc[m,n]
// Block = 256 threads = 8 waves. Each wave owns a 64x16 output strip
// (4 M-tiles sharing one B fragment per k-step -> 4 WMMAs per 5 loads).
// Block covers 64 rows x 128 cols. Requires M%64==0, N%128==0, K%4==0.
// ---------------------------------------------------------------------------
template <int TRANSB, bool GATED>
__global__ __launch_bounds__(256)
void gemm_wmma_kernel(const float* __restrict__ A, const float* __restrict__ Bm,
                      const float* __restrict__ bias, const float* __restrict__ Gsrc,
                      float* __restrict__ C,
                      int K, int lda, int ldb, int ldc,
                      long long strideA, long long strideB, long long strideC)
{
    const int wave = threadIdx.x >> 5;
    const int lane = threadIdx.x & 31;
    const int half = lane >> 4;     // 0: lanes 0-15, 1: lanes 16-31
    const int l    = lane & 15;

    const int m0 = blockIdx.x * 64;
    const int n0 = blockIdx.y * 128 + wave * 16;

    const float* Ab = A  + (long long)blockIdx.z * strideA;
    const float* Bb = Bm + (long long)blockIdx.z * strideB;
    float*       Cb = C  + (long long)blockIdx.z * strideC;

    // A fragment base: lane l holds A[row, k0+2*half .. +1]
    const float* Ap = Ab + (long long)(m0 + l) * lda + 2 * half;
    const long long astep = (long long)16 * lda;      // next M-tile

    // B fragment base
    const float* Bp = TRANSB ? (Bb + (long long)(n0 + l) * ldb + 2 * half)
                             : (Bb + (n0 + l));

    v8f acc0 = {0.f,0.f,0.f,0.f,0.f,0.f,0.f,0.f};
    v8f acc1 = acc0, acc2 = acc0, acc3 = acc0;

    for (int k0 = 0; k0 < K; k0 += 4) {
        v2f b;
        if (TRANSB) {
            b = *(const v2f*)(Bp + k0);
        } else {
            b.x = Bp[(long long)(k0 + 2 * half) * ldb];
            b.y = Bp[(long long)(k0 + 2 * half + 1) * ldb];
        }
        v2f a0 = *(const v2f*)(Ap + k0);
        v2f a1 = *(const v2f*)(Ap + astep + k0);
        v2f a2 = *(const v2f*)(Ap + 2 * astep + k0);
        v2f a3 = *(const v2f*)(Ap + 3 * astep + k0);
        acc0 = wmma4(a0, b, acc0);
        acc1 = wmma4(a1, b, acc1);
        acc2 = wmma4(a2, b, acc2);
        acc3 = wmma4(a3, b, acc3);
    }

    const int   col = n0 + l;
    const float bv  = bias ? bias[col] : 0.0f;

    v8f accs[4] = {acc0, acc1, acc2, acc3};
#pragma unroll
    for (int mi = 0; mi < 4; ++mi) {
#pragma unroll
        for (int r = 0; r < 8; ++r) {
            const int row = m0 + mi * 16 + r + half * 8;  // C layout: VGPR r -> M=r+8*half
            float cv = accs[mi][r] + bv;
            if (GATED) {
                cv = sigmoidf_(cv) * Gsrc[(long long)row * ldc + col];
            }
            Cb[(long long)row * ldc + col] = cv;
        }
    }
}

// ---------------------------------------------------------------------------
// Additive-attention scores + masked softmax.
//   attn[b,q,k] = softmax_k( v_attn . tanh(own[b,q,:] + comp[b,k,:]) , k<len[b] )
// One block (256 threads) per (q, b).
// ---------------------------------------------------------------------------
__global__ __launch_bounds__(256)
void attn_softmax_kernel(const float* __restrict__ own, const float* __restrict__ comp,
                         const float* __restrict__ v_attn, const int* __restrict__ lengths,
                         float* __restrict__ attn)
{
    const int q = blockIdx.x;
    const int b = blockIdx.y;
    const int tid = threadIdx.x;

    __shared__ float sOwn[H_];
    __shared__ float sVa[H_];
    __shared__ float sScore[L_];
    __shared__ float red[256];

    if (tid < H_) {
        sOwn[tid] = own[((long long)b * L_ + q) * H_ + tid];
        sVa[tid]  = v_attn[tid];
    }
    __syncthreads();

    const int len = lengths[b];
    for (int k = tid; k < L_; k += 256) {
        float s;
        if (k < len) {
            const float* cr = comp + ((long long)b * L_ + k) * H_;
            s = 0.0f;
#pragma unroll 4
            for (int h = 0; h < H_; ++h)
                s += sVa[h] * tanhf(sOwn[h] + cr[h]);
        } else {
            s = -1e30f;
        }
        sScore[k] = s;
    }
    __syncthreads();

    // block-wide max
    float m = -1e30f;
    for (int k = tid; k < L_; k += 256) m = fmaxf(m, sScore[k]);
    red[tid] = m;
    __syncthreads();
    for (int s = 128; s > 0; s >>= 1) {
        if (tid < s) red[tid] = fmaxf(red[tid], red[tid + s]);
        __syncthreads();
    }
    const float mx = red[0];
    __syncthreads();

    // exp + sum
    float psum = 0.0f;
    for (int k = tid; k < L_; k += 256) {
        float e = __expf(sScore[k] - mx);
        sScore[k] = e;
        psum += e;
    }
    red[tid] = psum;
    __syncthreads();
    for (int s = 128; s > 0; s >>= 1) {
        if (tid < s) red[tid] += red[tid + s];
        __syncthreads();
    }
    const float inv = 1.0f / red[0];

    float* arow = attn + ((long long)b * L_ + q) * L_;
    for (int k = tid; k < L_; k += 256) arow[k] = sScore[k] * inv;
}

// ---------------------------------------------------------------------------
// inp = concat(v, ctx) along features: [B*L, 2D]
// ---------------------------------------------------------------------------
__global__ __launch_bounds__(256)
void concat_kernel(const float* __restrict__ v, const float* __restrict__ ctx,
                   float* __restrict__ inp, int total)
{
    int i = blockIdx.x * blockDim.x + threadIdx.x;
    if (i >= total) return;
    int row = i >> 9;          // / (2D)
    int col = i & 511;         // % (2D)
    inp[i] = (col < D_) ? v[(long long)row * D_ + col]
                        : ctx[(long long)row * D_ + (col - D_)];
}

// ---------------------------------------------------------------------------
// Single-direction GRU. One block, 512 threads (16 waves).
//   hp = h @ w_hh^T     via WMMA on zero-padded 16x128 A tile in LDS
//                       (waves 0..7, 3 N-tiles of 16 each -> N=384)
//   gates/elementwise   by all 512 threads (b = tid/128, j = tid%128)
// xp: [B*L, 3H] precomputed input projections (bias included).
// out column offset = rev * H (fwd writes [:,0:H], bwd writes [:,H:2H]).
// ---------------------------------------------------------------------------
__global__ __launch_bounds__(512)
void gru_kernel(const float* __restrict__ xp, const float* __restrict__ w_hh,
                const float* __restrict__ b_hh, const int* __restrict__ lengths,
                float* __restrict__ out, int rev)
{
    __shared__ float hA[16][H_];      // A matrix: rows = batch (0..3 valid, 4..15 zero)
    __shared__ float hp[B_][3 * H_];  // recurrent projections for valid rows

    const int tid  = threadIdx.x;
    const int wave = tid >> 5;
    const int lane = tid & 31;
    const int half = lane >> 4;
    const int l    = lane & 15;

    for (int i = tid; i < 16 * H_; i += 512) (&hA[0][0])[i] = 0.0f;
    __syncthreads();

    const int b   = tid >> 7;      // 0..3
    const int j   = tid & 127;     // 0..127
    const int len = lengths[b];

    for (int step = 0; step < L_; ++step) {
        const int t = rev ? (L_ - 1 - step) : step;

        // --- hp = hA @ w_hh^T  (16 x 384), WMMA waves 0..7 ---
        if (wave < 8) {
#pragma unroll
            for (int nt = 0; nt < 3; ++nt) {
                const int n0 = (wave * 3 + nt) * 16;
                v8f acc = {0.f,0.f,0.f,0.f,0.f,0.f,0.f,0.f};
                for (int k0 = 0; k0 < H_; k0 += 4) {
                    v2f a = *(const v2f*)&hA[l][k0 + 2 * half];
                    v2f bb = *(const v2f*)(w_hh + (long long)(n0 + l) * H_ + k0 + 2 * half);
                    acc = wmma4(a, bb, acc);
                }
                if (half == 0) {    // rows 0..3 live in c[0..3] of lanes 0..15
#pragma unroll
                    for (int r = 0; r < B_; ++r) hp[r][n0 + l] = acc[r];
                }
            }
        }
        __syncthreads();

        // --- gates (all 512 threads) ---
        const float hprev = hA[b][j];
        const float* xrow = xp + ((long long)(b * L_ + t)) * (3 * H_);
        const float xr = xrow[j];
        const float xz = xrow[H_ + j];
        const float xn = xrow[2 * H_ + j];
        const float hr = hp[b][j]          + b_hh[j];
        const float hz = hp[b][H_ + j]     + b_hh[H_ + j];
        const float hn = hp[b][2 * H_ + j] + b_hh[2 * H_ + j];

        const float r = sigmoidf_(xr + hr);
        const float z = sigmoidf_(xz + hz);
        const float n = tanhf(xn + r * hn);
        const float hnew = (1.0f - z) * n + z * hprev;

        const bool msk = (t < len);
        hA[b][j] = msk ? hnew : hprev;
        out[((long long)(b * L_ + t)) * (2 * H_) + rev * H_ + j] = msk ? hnew : 0.0f;
        __syncthreads();
    }
}

// ---------------------------------------------------------------------------
extern "C" void kernel_launch(void* const* d_in, const int* in_sizes, int n_in,
                              void* d_out, int out_size, void* d_ws, size_t ws_size,
                              hipStream_t stream)
{
    (void)in_sizes; (void)n_in; (void)out_size; (void)ws_size;

    const float* v       = (const float*)d_in[0];
    const int*   lengths = (const int*)  d_in[1];
    // d_in[2] = p_mask (bool) -- recomputed from lengths instead
    const float* own_W   = (const float*)d_in[3];
    const float* own_b   = (const float*)d_in[4];
    const float* comp_W  = (const float*)d_in[5];
    const float* comp_b  = (const float*)d_in[6];
    const float* v_attn  = (const float*)d_in[7];
    const float* gate_W  = (const float*)d_in[8];
    const float* gate_b  = (const float*)d_in[9];
    const float* w_ih_f  = (const float*)d_in[10];
    const float* w_hh_f  = (const float*)d_in[11];
    const float* b_ih_f  = (const float*)d_in[12];
    const float* b_hh_f  = (const float*)d_in[13];
    const float* w_ih_b  = (const float*)d_in[14];
    const float* w_hh_b  = (const float*)d_in[15];
    const float* b_ih_b  = (const float*)d_in[16];
    const float* b_hh_b  = (const float*)d_in[17];
    float* out = (float*)d_out;

    // Workspace layout (floats)
    const int M = B_ * L_;                       // 2048 rows
    float* ws    = (float*)d_ws;
    float* own   = ws;                           // [M, H]      262144
    float* comp  = own  + (size_t)M * H_;        // [M, H]      262144
    float* attn  = comp + (size_t)M * H_;        // [B, L, L]  1048576
    float* ctx   = attn + (size_t)B_ * L_ * L_;  // [M, D]      524288
    float* inp   = ctx  + (size_t)M * D_;        // [M, 2D]    1048576
    float* inpg  = inp  + (size_t)M * 2 * D_;    // [M, 2D]    1048576
    float* xp_f  = inpg + (size_t)M * 2 * D_;    // [M, 3H]     786432
    float* xp_b  = xp_f + (size_t)M * 3 * H_;    // [M, 3H]     786432

    // 1-2: own/comp projections  (M=2048, N=128, K=256, C = v * W^T + b)
    {
        dim3 grid(M / 64, H_ / 128, 1);
        gemm_wmma_kernel<1, false><<<grid, 256, 0, stream>>>(v, own_W, own_b, nullptr, own,
            D_, D_, D_, H_, 0, 0, 0);
        gemm_wmma_kernel<1, false><<<grid, 256, 0, stream>>>(v, comp_W, comp_b, nullptr, comp,
            D_, D_, D_, H_, 0, 0, 0);
    }

    // 3: additive-attention scores + masked softmax -> attn [B,L,L]
    {
        dim3 grid(L_, B_, 1);
        attn_softmax_kernel<<<grid, 256, 0, stream>>>(own, comp, v_attn, lengths, attn);
    }

    // 4: ctx[b] = attn[b] @ v[b]   (batched: M=512, N=256, K=512)
    {
        dim3 grid(L_ / 64, D_ / 128, B_);
        gemm_wmma_kernel<0, false><<<grid, 256, 0, stream>>>(attn, v, nullptr, nullptr, ctx,
            L_, L_, D_, D_,
            (long long)L_ * L_, (long long)L_ * D_, (long long)L_ * D_);
    }

    // 5: inp = concat(v, ctx)
    {
        const int total = M * 2 * D_;
        concat_kernel<<<(total + 255) / 256, 256, 0, stream>>>(v, ctx, inp, total);
    }

    // 6: inpg = sigmoid(inp @ gate_W^T + gate_b) * inp   (M=2048, N=512, K=512)
    {
        dim3 grid(M / 64, (2 * D_) / 128, 1);
        gemm_wmma_kernel<1, true><<<grid, 256, 0, stream>>>(inp, gate_W, gate_b, inp, inpg,
            2 * D_, 2 * D_, 2 * D_, 2 * D_, 0, 0, 0);
    }

    // 7: GRU input projections xp = inpg @ w_ih^T + b_ih  (M=2048, N=384, K=512)
    {
        dim3 grid(M / 64, (3 * H_) / 128, 1);
        gemm_wmma_kernel<1, false><<<grid, 256, 0, stream>>>(inpg, w_ih_f, b_ih_f, nullptr, xp_f,
            2 * D_, 2 * D_, 2 * D_, 3 * H_, 0, 0, 0);
        gemm_wmma_kernel<1, false><<<grid, 256, 0, stream>>>(inpg, w_ih_b, b_ih_b, nullptr, xp_b,
            2 * D_, 2 * D_, 2 * D_, 3 * H_, 0, 0, 0);
    }

    // 8-9: bidirectional GRU recurrence (WMMA on padded 16x128 hidden state)
    gru_kernel<<<1, 512, 0, stream>>>(xp_f, w_hh_f, b_hh_f, lengths, out, /*rev=*/0);
    gru_kernel<<<1, 512, 0, stream>>>(xp_b, w_hh_b, b_hh_b, lengths, out, /*rev=*/1);
}